// GELU68_17566416240654
// MI455X (gfx1250) — compile-verified
//
#include <hip/hip_runtime.h>
#include <math.h>

// ---------------- constants from the reference ----------------
#define U_VAL     0.3f
#define REC_RATE  0.221199216928595f   // 1 - exp(-1/4)
#define EPS_MEAN  1e-5f
#define NORM_EPS  1e-12f
#define R_CLIP_LO 0.01f
#define R_CLIP_HI 1.0f

#define BB 8
#define TT 4096
#define DD 1024
#define ROWS (BB*TT)            // 32768

// workspace layout (float offsets)
#define W_OFF   0               // 1024 floats: normalized ema_out
#define MU_OFF  1024            // 1 float: mean(ema_resource)
#define FAM_OFF 2048            // 32768 floats: familiarity
#define AB_OFF  36864           // 32768 float2: exclusive affine prefix (A_t, B_t)

typedef __attribute__((ext_vector_type(2))) float v2f;
typedef __attribute__((ext_vector_type(8))) float v8f;

// 0.5*x*(1 + tanh(c*(x + 0.044715 x^3)))  ==  x * sigmoid(2c*(x+0.044715x^3))
// Branch-free: one v_exp_f32 + one v_rcp_f32 (TRANS ops co-execute with WMMA);
// avoids the IEEE div_fmas/div_fixup sequence and its VCC traffic.
__device__ __forceinline__ float gelu_tanh(float x) {
  const float c2 = 1.5957691216057308f;            // 2*sqrt(2/pi)
  float x3 = x * x * x;
  float u2 = c2 * fmaf(0.044715f, x3, x);
  float e  = __expf(-u2);
  return x * __builtin_amdgcn_rcpf(1.0f + e);
}

// ---------------- kernel 0: normalize ema_out, mean(ema_resource) ----------------
__global__ void prep_kernel(const float* __restrict__ ema_resource,
                            const float* __restrict__ ema_out,
                            float* __restrict__ ws) {
  __shared__ float red[1024];
  int t = threadIdx.x;
  float v = ema_out[t];
  red[t] = v * v;
  __syncthreads();
  for (int s = 512; s > 0; s >>= 1) {
    if (t < s) red[t] += red[t + s];
    __syncthreads();
  }
  float inv = 1.0f / fmaxf(sqrtf(red[0]), NORM_EPS);
  __syncthreads();
  ws[W_OFF + t] = v * inv;

  float r = ema_resource[t];
  red[t] = r;
  __syncthreads();
  for (int s = 512; s > 0; s >>= 1) {
    if (t < s) red[t] += red[t + s];
    __syncthreads();
  }
  if (t == 0) ws[MU_OFF] = red[0] * (1.0f / 1024.0f);
}

// ---------------- kernel 1: familiarity via WMMA row-dot + VALU row-norm ----------
// One wave handles 16 rows. A-tile (f32 16x4): lane L (<16) holds row L,
// k = 4j+{0,1}; lane L+16 holds row L, k = 4j+{2,3}. B (4x16) = w[k] replicated
// in all 16 columns -> every column of C holds the per-row dot.
__global__ void fam_kernel(const float* __restrict__ x,
                           const float* __restrict__ ws,
                           float* __restrict__ fam) {
  __shared__ float wlds[DD];
  int tid = threadIdx.x;
  // preload normalized ema vector (256 threads x 4)
  *(float4*)&wlds[tid * 4] = *(const float4*)&ws[W_OFF + tid * 4];
  __syncthreads();

  int lane  = tid & 31;
  int wv    = tid >> 5;
  int rowb  = (blockIdx.x * 8 + wv) * 16;           // 16 rows per wave
  int mrow  = lane & 15;
  int khalf = (lane >> 4) << 1;                     // 0 for lanes 0-15, 2 for 16-31

  const float* xrow = x + (size_t)(rowb + mrow) * DD + khalf;

  v8f  c  = {};
  float ss = 0.0f;

  // software pipeline: keep the next iteration's load in flight
  float2 av = *(const float2*)(xrow);

#pragma unroll 4
  for (int j = 0; j < DD / 4; ++j) {
    int jn = (j + 1 < DD / 4) ? (j + 1) : j;        // clamped prefetch (in-bounds)
    float2 av_next = *(const float2*)(xrow + jn * 4);

    float a0 = gelu_tanh(av.x);
    float a1 = gelu_tanh(av.y);
    ss = fmaf(a0, a0, ss);
    ss = fmaf(a1, a1, ss);
    float2 wv2 = *(const float2*)&wlds[j * 4 + khalf];
    v2f A; A[0] = a0;     A[1] = a1;
    v2f B; B[0] = wv2.x;  B[1] = wv2.y;
    // D = A(16x4) * B(4x16) + C ; all 16 columns identical = per-row dot
    c = __builtin_amdgcn_wmma_f32_16x16x4_f32(false, A, false, B,
                                              (short)0, c, false, false);
    av = av_next;
  }

  // full ||out||^2 for row (lane&15): combine even/odd-pair halves
  ss += __shfl_xor(ss, 16, 32);
  float inv_norm = 1.0f / fmaxf(sqrtf(ss), NORM_EPS);

  // C layout: VGPR v, lanes 0-15 -> row v ; lanes 16-31 -> row v+8.
  // Lane v writes row v; lane 24+v (its ss corresponds to row v+8) writes row v+8.
#pragma unroll
  for (int v = 0; v < 8; ++v) {
    float dot = c[v];
    if (lane == v) {
      float cs = dot * inv_norm;
      fam[rowb + v] = fminf(fmaxf(cs, 0.0f), 1.0f);   // relu(clip(cos,-1,1))
    }
    if (lane == 24 + v) {
      float cs = dot * inv_norm;
      fam[rowb + v + 8] = fminf(fmaxf(cs, 0.0f), 1.0f);
    }
  }
}

// ---------------- kernel 2: parallel scan of affine maps over T ------------------
// r_{t+1} = a_t r_t + REC_RATE, a_t = 1 - U*fam_t - REC_RATE (clip provably
// inactive for these constants).  Exclusive prefix (A_t,B_t): r_t = A_t r0 + B_t.
__global__ void scan_kernel(const float* __restrict__ fam,
                            float2* __restrict__ ab) {
  __shared__ float2 sc[256];
  int b = blockIdx.x, tid = threadIdx.x;

  float fv[16];
  const float* f = fam + b * TT + tid * 16;
#pragma unroll
  for (int k = 0; k < 16; ++k) fv[k] = f[k];

  // compose this thread's 16-step chunk (apply in increasing t)
  float A = 1.0f, Bc = 0.0f;
#pragma unroll
  for (int k = 0; k < 16; ++k) {
    float a = 1.0f - U_VAL * fv[k] - REC_RATE;
    A  = a * A;
    Bc = fmaf(a, Bc, REC_RATE);
  }
  float2 val = make_float2(A, Bc);
  sc[tid] = val;
  __syncthreads();

  // Hillis-Steele inclusive scan, combine(left,right) = right ∘ left
  for (int off = 1; off < 256; off <<= 1) {
    float2 left;
    bool has = tid >= off;
    if (has) left = sc[tid - off];
    __syncthreads();
    if (has) {
      float nA = val.x * left.x;
      val.y = fmaf(val.x, left.y, val.y);
      val.x = nA;
      sc[tid] = val;
    }
    __syncthreads();
  }

  float exA = 1.0f, exB = 0.0f;
  if (tid > 0) { float2 p = sc[tid - 1]; exA = p.x; exB = p.y; }

  float2* out = ab + b * TT + tid * 16;
#pragma unroll
  for (int k = 0; k < 16; ++k) {
    out[k] = make_float2(exA, exB);                  // state BEFORE step t
    float a = 1.0f - U_VAL * fv[k] - REC_RATE;
    float nA = a * exA;
    exB = fmaf(a, exB, REC_RATE);
    exA = nA;
  }
}

// ---------------- kernel 3: fused gelu * r / (mean_d r + eps) --------------------
__global__ void out_kernel(const float* __restrict__ x,
                           const float* __restrict__ r0,
                           const float* __restrict__ ws,
                           const float2* __restrict__ ab,
                           float* __restrict__ out) {
  int row = blockIdx.x;
  int t = row & (TT - 1);
  float2 AB = ab[row];
  float mu0 = ws[MU_OFF];
  float inv_den = 1.0f / (fmaf(AB.x, mu0, AB.y) + EPS_MEAN);

  int d0 = threadIdx.x * 4;
  float4 xv = *(const float4*)(x + (size_t)row * DD + d0);
  float4 rv = *(const float4*)(r0 + d0);

  float g0 = gelu_tanh(xv.x), g1 = gelu_tanh(xv.y);
  float g2 = gelu_tanh(xv.z), g3 = gelu_tanh(xv.w);

  float r0v, r1v, r2v, r3v;
  if (t == 0) {                                      // trace emits raw r0 at t=0
    r0v = rv.x; r1v = rv.y; r2v = rv.z; r3v = rv.w;
  } else {
    r0v = fminf(fmaxf(fmaf(AB.x, rv.x, AB.y), R_CLIP_LO), R_CLIP_HI);
    r1v = fminf(fmaxf(fmaf(AB.x, rv.y, AB.y), R_CLIP_LO), R_CLIP_HI);
    r2v = fminf(fmaxf(fmaf(AB.x, rv.z, AB.y), R_CLIP_LO), R_CLIP_HI);
    r3v = fminf(fmaxf(fmaf(AB.x, rv.w, AB.y), R_CLIP_LO), R_CLIP_HI);
  }

  float4 o;
  o.x = g0 * r0v * inv_den;
  o.y = g1 * r1v * inv_den;
  o.z = g2 * r2v * inv_den;
  o.w = g3 * r3v * inv_den;
  *(float4*)(out + (size_t)row * DD + d0) = o;
}

// ---------------- launcher ----------------
extern "C" void kernel_launch(void* const* d_in, const int* in_sizes, int n_in,
                              void* d_out, int out_size, void* d_ws, size_t ws_size,
                              hipStream_t stream) {
  const float* x            = (const float*)d_in[0];
  const float* ema_resource = (const float*)d_in[1];
  const float* ema_out      = (const float*)d_in[2];
  float* outp = (float*)d_out;
  float* ws   = (float*)d_ws;
  float* fam  = ws + FAM_OFF;
  float2* ab  = (float2*)(ws + AB_OFF);

  prep_kernel<<<1, 1024, 0, stream>>>(ema_resource, ema_out, ws);
  fam_kernel<<<ROWS / (16 * 8), 256, 0, stream>>>(x, ws, fam);   // 256 blocks
  scan_kernel<<<BB, 256, 0, stream>>>(fam, ab);
  out_kernel<<<ROWS, 256, 0, stream>>>(x, ema_resource, ws, ab, outp);
}